// Maploss_v3_3358664425475
// MI455X (gfx1250) — compile-verified
//
#include <hip/hip_runtime.h>
#include <hip/hip_bf16.h>
#include <stdint.h>

// ---------------------------------------------------------------------------
// Maploss (CRAFT-style OHEM loss) for MI455X / gfx1250.
// B=64 images, 512x512, 2 channels -> 128 independent top-k selection problems.
// Exact top-k sum via 3-pass radix select (11+11+10 bits) on non-negative f32
// bit patterns; optional value-cache in workspace to cut HBM traffic ~2x.
// ---------------------------------------------------------------------------

#define BATCH          64
#define NPIX           (512 * 512)       // 262144 pixels per image
#define BLOCKS_PER_IMG 256
#define PIX_PER_BLOCK  1024              // 256 threads x 4 contiguous pixels
#define NPROB          128               // 2 channels * 64 images
#define POS_THRESH     0.1f
#define FALLBACK_POS   1000.0f
#define BINS0          2048              // bits [31:21]
#define BINS1          2048              // bits [20:10]
#define BINS2          1024              // bits [ 9: 0]

typedef __attribute__((ext_vector_type(2))) float v2f;
typedef __attribute__((ext_vector_type(8))) float v8f;

// --- fixed-order block reduction (deterministic) ---------------------------
__device__ __forceinline__ float blk_reduce_sum256(float v, float* red) {
    const int t = threadIdx.x;
    red[t] = v;
    __syncthreads();
    #pragma unroll
    for (int s = 128; s > 0; s >>= 1) {
        if (t < s) red[t] += red[t + s];
        __syncthreads();
    }
    float r = red[0];
    __syncthreads();
    return r;
}

__device__ __forceinline__ float neg_val(float lab, float pre, float m,
                                         bool& pos, float& loss) {
    float d = pre - lab;
    loss = d * d * m;                // masked per-pixel MSE
    pos  = lab > POS_THRESH;
    return pos ? 0.0f : loss;        // element of neg_region (pos pixels -> 0)
}

// ---------------------------------------------------------------------------
// Pass 1: fused stats (p, pos_sum, neg_sum) + 2048-bin top-digit histogram.
// Optionally spills neg-value bit patterns to ws (value cache).
// ---------------------------------------------------------------------------
__global__ void k_stats_hist0(const float* __restrict__ lab_r,
                              const float* __restrict__ lab_a,
                              const float* __restrict__ pre_r,
                              const float* __restrict__ pre_a,
                              const float* __restrict__ mask,
                              uint32_t* __restrict__ hist0,
                              float* __restrict__ part_pcnt,
                              float* __restrict__ part_pos,
                              float* __restrict__ part_neg,
                              uint32_t* __restrict__ vals_r,
                              uint32_t* __restrict__ vals_a,
                              int cache) {
    __shared__ uint32_t hist[2][BINS0];
    __shared__ float red[256];
    const int t   = threadIdx.x;
    const int img = blockIdx.x / BLOCKS_PER_IMG;
    const int blk = blockIdx.x % BLOCKS_PER_IMG;
    for (int b = t; b < BINS0; b += 256) { hist[0][b] = 0u; hist[1][b] = 0u; }
    __syncthreads();

    const int g = img * NPIX + blk * PIX_PER_BLOCK + t * 4;
    const float4 m4  = *(const float4*)&mask[g];
    const float4 l0  = *(const float4*)&lab_r[g];
    const float4 p0  = *(const float4*)&pre_r[g];
    const float4 l1  = *(const float4*)&lab_a[g];
    const float4 p1  = *(const float4*)&pre_a[g];
    const float* mm = (const float*)&m4;
    const float* L0 = (const float*)&l0;  const float* P0 = (const float*)&p0;
    const float* L1 = (const float*)&l1;  const float* P1 = (const float*)&p1;

    float pc[2] = {0.f, 0.f}, ps[2] = {0.f, 0.f}, ns[2] = {0.f, 0.f};
    uint32_t vb0[4], vb1[4];
    #pragma unroll
    for (int j = 0; j < 4; ++j) {
        {
            bool pos; float loss;
            float v = neg_val(L0[j], P0[j], mm[j], pos, loss);
            pc[0] += pos ? 1.f : 0.f;
            ps[0] += pos ? loss : 0.f;
            ns[0] += v;
            vb0[j] = __float_as_uint(v);
            atomicAdd(&hist[0][vb0[j] >> 21], 1u);
        }
        {
            bool pos; float loss;
            float v = neg_val(L1[j], P1[j], mm[j], pos, loss);
            pc[1] += pos ? 1.f : 0.f;
            ps[1] += pos ? loss : 0.f;
            ns[1] += v;
            vb1[j] = __float_as_uint(v);
            atomicAdd(&hist[1][vb1[j] >> 21], 1u);
        }
    }
    if (cache) {
        uint4 u0; u0.x = vb0[0]; u0.y = vb0[1]; u0.z = vb0[2]; u0.w = vb0[3];
        uint4 u1; u1.x = vb1[0]; u1.y = vb1[1]; u1.z = vb1[2]; u1.w = vb1[3];
        *(uint4*)&vals_r[g] = u0;
        *(uint4*)&vals_a[g] = u1;
    }
    __syncthreads();
    for (int b = t; b < BINS0; b += 256) {
        uint32_t c0 = hist[0][b];
        if (c0) atomicAdd(&hist0[img * BINS0 + b], c0);
        uint32_t c1 = hist[1][b];
        if (c1) atomicAdd(&hist0[(BATCH + img) * BINS0 + b], c1);
    }

    #pragma unroll
    for (int ch = 0; ch < 2; ++ch) {
        const int prob = ch * BATCH + img;
        const int idx  = prob * BLOCKS_PER_IMG + blk;
        float r;
        r = blk_reduce_sum256(pc[ch], red); if (t == 0) part_pcnt[idx] = r;
        r = blk_reduce_sum256(ps[ch], red); if (t == 0) part_pos[idx]  = r;
        r = blk_reduce_sum256(ns[ch], red); if (t == 0) part_neg[idx]  = r;
    }
}

// ---------------------------------------------------------------------------
// WMMA f32 reduction: sum 256 block-partials per problem on the matrix pipe.
// D = A x ones + C accumulates row sums; total = s(lane0) + s(lane16).
// One full wave32 per block => EXEC all ones at the WMMA, as required.
// ---------------------------------------------------------------------------
__global__ void k_reduce256_wmma(const float* __restrict__ part,
                                 float* __restrict__ out) {
    const int prob = blockIdx.x;
    const int lane = threadIdx.x;            // 0..31
    const float* src = part + prob * 256;

    v8f acc = {};
    v2f b;  b.x = 1.0f;  b.y = 1.0f;         // all-ones B (4x16)
    #pragma unroll
    for (int i = 0; i < 4; ++i) {
        v2f a;
        a.x = src[i * 64 + lane];
        a.y = src[i * 64 + 32 + lane];
        acc = __builtin_amdgcn_wmma_f32_16x16x4_f32(
            false, a, false, b, (short)0, acc, false, false);
    }
    float s = acc[0] + acc[1] + acc[2] + acc[3] +
              acc[4] + acc[5] + acc[6] + acc[7];
    float total = __shfl(s, 0, 32) + __shfl(s, 16, 32);
    if (lane == 0) out[prob] = total;
}

// ---------------------------------------------------------------------------
// Prep: k = floor(neg_rto * p_eff); kk = clamp(k, 1, NPIX); prefix = 0
// ---------------------------------------------------------------------------
__global__ void k_prep(const float* __restrict__ pcnt,
                       const int* __restrict__ neg_rto,
                       int* __restrict__ kvals,
                       uint32_t* __restrict__ kk,
                       uint32_t* __restrict__ prefix) {
    const int p = threadIdx.x;
    if (p >= NPROB) return;
    const float rto   = (float)(*neg_rto);
    const float pv    = pcnt[p];
    const float p_eff = (pv > 0.f) ? pv : FALLBACK_POS;
    const int   k     = (int)floorf(rto * p_eff);
    kvals[p]  = k;
    int kc = k; if (kc < 1) kc = 1; if (kc > NPIX) kc = NPIX;
    kk[p]     = (uint32_t)kc;
    prefix[p] = 0u;
}

// ---------------------------------------------------------------------------
// Radix-select step: pick next digit of the k-th largest value (desc scan).
// ---------------------------------------------------------------------------
__global__ void k_select(const uint32_t* __restrict__ hist, uint32_t nbins,
                         uint32_t shift,
                         uint32_t* __restrict__ prefix,
                         uint32_t* __restrict__ kk) {
    const int p = threadIdx.x;
    if (p >= NPROB) return;
    const uint32_t* h = hist + (uint32_t)p * nbins;
    const uint32_t k = kk[p];
    uint32_t cum = 0, chosen = 0, kn = 1;
    bool found = false;
    for (int b = (int)nbins - 1; b >= 0; --b) {
        const uint32_t c = h[b];
        if (k <= cum + c) { chosen = (uint32_t)b; kn = k - cum; found = true; break; }
        cum += c;
    }
    if (!found) { chosen = 0u; kn = 1u; }    // k > total: overridden in finalize
    prefix[p] |= (chosen << shift);
    kk[p] = kn;
}

// ---------------------------------------------------------------------------
// Refinement histogram pass: bin next digit among prefix-matching values.
// Reads cached bit patterns if available, else recomputes from inputs.
// ---------------------------------------------------------------------------
__global__ void k_hist_pass(const float* __restrict__ lab_r,
                            const float* __restrict__ lab_a,
                            const float* __restrict__ pre_r,
                            const float* __restrict__ pre_a,
                            const float* __restrict__ mask,
                            const uint32_t* __restrict__ vals_r,
                            const uint32_t* __restrict__ vals_a,
                            int cache,
                            const uint32_t* __restrict__ prefix,
                            uint32_t shift, uint32_t matchshift,
                            uint32_t* __restrict__ hist, uint32_t nbins) {
    __shared__ uint32_t h[2][BINS0];         // max bins
    const int t   = threadIdx.x;
    const int img = blockIdx.x / BLOCKS_PER_IMG;
    const int blk = blockIdx.x % BLOCKS_PER_IMG;
    for (int b = t; b < (int)nbins; b += 256) { h[0][b] = 0u; h[1][b] = 0u; }
    __syncthreads();

    const uint32_t pref0 = prefix[img];
    const uint32_t pref1 = prefix[BATCH + img];
    const uint32_t mask_bins = nbins - 1u;
    const int g = img * NPIX + blk * PIX_PER_BLOCK + t * 4;

    uint32_t b0[4], b1[4];
    if (cache) {
        const uint4 u0 = *(const uint4*)&vals_r[g];
        const uint4 u1 = *(const uint4*)&vals_a[g];
        b0[0] = u0.x; b0[1] = u0.y; b0[2] = u0.z; b0[3] = u0.w;
        b1[0] = u1.x; b1[1] = u1.y; b1[2] = u1.z; b1[3] = u1.w;
    } else {
        const float4 m4 = *(const float4*)&mask[g];
        const float4 l0 = *(const float4*)&lab_r[g];
        const float4 p0 = *(const float4*)&pre_r[g];
        const float4 l1 = *(const float4*)&lab_a[g];
        const float4 p1 = *(const float4*)&pre_a[g];
        const float* mm = (const float*)&m4;
        const float* L0 = (const float*)&l0;  const float* P0 = (const float*)&p0;
        const float* L1 = (const float*)&l1;  const float* P1 = (const float*)&p1;
        #pragma unroll
        for (int j = 0; j < 4; ++j) {
            bool pos; float loss;
            b0[j] = __float_as_uint(neg_val(L0[j], P0[j], mm[j], pos, loss));
            b1[j] = __float_as_uint(neg_val(L1[j], P1[j], mm[j], pos, loss));
        }
    }
    #pragma unroll
    for (int j = 0; j < 4; ++j) {
        if (((b0[j] ^ pref0) >> matchshift) == 0u)
            atomicAdd(&h[0][(b0[j] >> shift) & mask_bins], 1u);
        if (((b1[j] ^ pref1) >> matchshift) == 0u)
            atomicAdd(&h[1][(b1[j] >> shift) & mask_bins], 1u);
    }
    __syncthreads();
    for (int b = t; b < (int)nbins; b += 256) {
        uint32_t c0 = h[0][b];
        if (c0) atomicAdd(&hist[img * nbins + b], c0);
        uint32_t c1 = h[1][b];
        if (c1) atomicAdd(&hist[(BATCH + img) * nbins + b], c1);
    }
}

// ---------------------------------------------------------------------------
// Final sweep: sum of values strictly greater than v_k (full 32-bit compare).
// ---------------------------------------------------------------------------
__global__ void k_sum_greater(const float* __restrict__ lab_r,
                              const float* __restrict__ lab_a,
                              const float* __restrict__ pre_r,
                              const float* __restrict__ pre_a,
                              const float* __restrict__ mask,
                              const uint32_t* __restrict__ vals_r,
                              const uint32_t* __restrict__ vals_a,
                              int cache,
                              const uint32_t* __restrict__ prefix,
                              float* __restrict__ part_sg) {
    __shared__ float red[256];
    const int t   = threadIdx.x;
    const int img = blockIdx.x / BLOCKS_PER_IMG;
    const int blk = blockIdx.x % BLOCKS_PER_IMG;
    const uint32_t thr0 = prefix[img];
    const uint32_t thr1 = prefix[BATCH + img];
    const int g = img * NPIX + blk * PIX_PER_BLOCK + t * 4;

    uint32_t b0[4], b1[4];
    if (cache) {
        const uint4 u0 = *(const uint4*)&vals_r[g];
        const uint4 u1 = *(const uint4*)&vals_a[g];
        b0[0] = u0.x; b0[1] = u0.y; b0[2] = u0.z; b0[3] = u0.w;
        b1[0] = u1.x; b1[1] = u1.y; b1[2] = u1.z; b1[3] = u1.w;
    } else {
        const float4 m4 = *(const float4*)&mask[g];
        const float4 l0 = *(const float4*)&lab_r[g];
        const float4 p0 = *(const float4*)&pre_r[g];
        const float4 l1 = *(const float4*)&lab_a[g];
        const float4 p1 = *(const float4*)&pre_a[g];
        const float* mm = (const float*)&m4;
        const float* L0 = (const float*)&l0;  const float* P0 = (const float*)&p0;
        const float* L1 = (const float*)&l1;  const float* P1 = (const float*)&p1;
        #pragma unroll
        for (int j = 0; j < 4; ++j) {
            bool pos; float loss;
            b0[j] = __float_as_uint(neg_val(L0[j], P0[j], mm[j], pos, loss));
            b1[j] = __float_as_uint(neg_val(L1[j], P1[j], mm[j], pos, loss));
        }
    }
    float sg[2] = {0.f, 0.f};
    #pragma unroll
    for (int j = 0; j < 4; ++j) {
        if (b0[j] > thr0) sg[0] += __uint_as_float(b0[j]);
        if (b1[j] > thr1) sg[1] += __uint_as_float(b1[j]);
    }
    #pragma unroll
    for (int ch = 0; ch < 2; ++ch) {
        const int prob = ch * BATCH + img;
        float r = blk_reduce_sum256(sg[ch], red);
        if (t == 0) part_sg[prob * BLOCKS_PER_IMG + blk] = r;
    }
}

// ---------------------------------------------------------------------------
// Finalize: per-problem loss, deterministic 128-way tree sum, /BATCH.
// ---------------------------------------------------------------------------
__global__ void k_finalize(const float* __restrict__ pcnt,
                           const float* __restrict__ pos_sum,
                           const float* __restrict__ neg_sum,
                           const float* __restrict__ sum_g,
                           const uint32_t* __restrict__ prefix,
                           const uint32_t* __restrict__ kk,
                           const int* __restrict__ kvals,
                           const int* __restrict__ neg_rto,
                           float* __restrict__ out) {
    __shared__ float red[NPROB];
    const int p = threadIdx.x;
    float loss = 0.f;
    if (p < NPROB) {
        const float rto   = (float)(*neg_rto);
        const float pv    = pcnt[p];
        const float posl  = (pv > 0.f) ? (pos_sum[p] / fmaxf(pv, 1.f)) : 0.f;
        const float p_eff = (pv > 0.f) ? pv : FALLBACK_POS;
        const float k_f   = rto * p_eff;
        const int   k     = kvals[p];
        const float nneg  = (float)NPIX - pv;
        const float vk    = __uint_as_float(prefix[p]);
        const float topk  = (k > NPIX) ? neg_sum[p]
                                       : (sum_g[p] + (float)kk[p] * vk);
        const bool use_all = (pv > 0.f) && (nneg < k_f);
        const float negl = use_all ? (neg_sum[p] / fmaxf(nneg, 1.f))
                                   : (topk / k_f);
        loss = posl + negl;
    }
    red[p] = loss;
    __syncthreads();
    #pragma unroll
    for (int s = 64; s > 0; s >>= 1) {
        if (p < s) red[p] += red[p + s];
        __syncthreads();
    }
    if (p == 0) out[0] = red[0] / (float)BATCH;
}

// ---------------------------------------------------------------------------
// Host launcher
// ---------------------------------------------------------------------------
extern "C" void kernel_launch(void* const* d_in, const int* in_sizes, int n_in,
                              void* d_out, int out_size, void* d_ws, size_t ws_size,
                              hipStream_t stream) {
    (void)in_sizes; (void)n_in; (void)out_size;
    const float* lab_r   = (const float*)d_in[0];
    const float* lab_a   = (const float*)d_in[1];
    const float* pre_r   = (const float*)d_in[2];
    const float* pre_a   = (const float*)d_in[3];
    const float* mask    = (const float*)d_in[4];
    const int*   neg_rto = (const int*)d_in[5];
    float* out = (float*)d_out;

    // Workspace layout
    uint8_t* ws = (uint8_t*)d_ws;
    const size_t SZ0 = (size_t)NPROB * BINS0 * 4;     // 1 MB
    const size_t SZ1 = (size_t)NPROB * BINS1 * 4;     // 1 MB
    const size_t SZ2 = (size_t)NPROB * BINS2 * 4;     // 0.5 MB
    const size_t PB  = (size_t)NPROB * BLOCKS_PER_IMG * 4;   // 128 KB each
    uint32_t* hist0 = (uint32_t*)(ws);
    uint32_t* hist1 = (uint32_t*)(ws + SZ0);
    uint32_t* hist2 = (uint32_t*)(ws + SZ0 + SZ1);
    size_t off = SZ0 + SZ1 + SZ2;
    float* part_pcnt = (float*)(ws + off); off += PB;
    float* part_pos  = (float*)(ws + off); off += PB;
    float* part_neg  = (float*)(ws + off); off += PB;
    float* part_sg   = (float*)(ws + off); off += PB;
    float*    stats_pcnt = (float*)(ws + off); off += NPROB * 4;
    float*    stats_pos  = (float*)(ws + off); off += NPROB * 4;
    float*    stats_neg  = (float*)(ws + off); off += NPROB * 4;
    float*    stats_sg   = (float*)(ws + off); off += NPROB * 4;
    uint32_t* prefix     = (uint32_t*)(ws + off); off += NPROB * 4;
    uint32_t* kk         = (uint32_t*)(ws + off); off += NPROB * 4;
    int*      kvals      = (int*)(ws + off); off += NPROB * 4;
    off = (off + 255) & ~(size_t)255;                  // align value cache
    const size_t VALSZ = (size_t)BATCH * NPIX * 4;     // 64 MB per channel
    uint32_t* vals_r = (uint32_t*)(ws + off);
    uint32_t* vals_a = (uint32_t*)(ws + off + VALSZ);
    const int cache = (ws_size >= off + 2 * VALSZ) ? 1 : 0;

    const int data_blocks = BATCH * BLOCKS_PER_IMG;    // 16384

    hipMemsetAsync(hist0, 0, SZ0 + SZ1 + SZ2, stream); // zero all histograms

    k_stats_hist0<<<data_blocks, 256, 0, stream>>>(
        lab_r, lab_a, pre_r, pre_a, mask, hist0,
        part_pcnt, part_pos, part_neg, vals_r, vals_a, cache);

    k_reduce256_wmma<<<NPROB, 32, 0, stream>>>(part_pcnt, stats_pcnt);
    k_reduce256_wmma<<<NPROB, 32, 0, stream>>>(part_pos,  stats_pos);
    k_reduce256_wmma<<<NPROB, 32, 0, stream>>>(part_neg,  stats_neg);

    k_prep<<<1, NPROB, 0, stream>>>(stats_pcnt, neg_rto, kvals, kk, prefix);

    // digit 0: bits [31:21]
    k_select<<<1, NPROB, 0, stream>>>(hist0, BINS0, 21u, prefix, kk);
    // digit 1: bits [20:10], match on bits [31:21]
    k_hist_pass<<<data_blocks, 256, 0, stream>>>(
        lab_r, lab_a, pre_r, pre_a, mask, vals_r, vals_a, cache,
        prefix, 10u, 21u, hist1, BINS1);
    k_select<<<1, NPROB, 0, stream>>>(hist1, BINS1, 10u, prefix, kk);
    // digit 2: bits [9:0], match on bits [31:10]
    k_hist_pass<<<data_blocks, 256, 0, stream>>>(
        lab_r, lab_a, pre_r, pre_a, mask, vals_r, vals_a, cache,
        prefix, 0u, 10u, hist2, BINS2);
    k_select<<<1, NPROB, 0, stream>>>(hist2, BINS2, 0u, prefix, kk);

    k_sum_greater<<<data_blocks, 256, 0, stream>>>(
        lab_r, lab_a, pre_r, pre_a, mask, vals_r, vals_a, cache,
        prefix, part_sg);
    k_reduce256_wmma<<<NPROB, 32, 0, stream>>>(part_sg, stats_sg);

    k_finalize<<<1, NPROB, 0, stream>>>(
        stats_pcnt, stats_pos, stats_neg, stats_sg,
        prefix, kk, kvals, neg_rto, out);
}